// ModernHopfieldHead_61950608277571
// MI455X (gfx1250) — compile-verified
//
#include <hip/hip_runtime.h>
#include <math.h>

typedef float v2f __attribute__((ext_vector_type(2)));
typedef float v8f __attribute__((ext_vector_type(8)));
typedef unsigned v4u __attribute__((ext_vector_type(4)));
typedef int v4i __attribute__((ext_vector_type(4)));
typedef int v8i __attribute__((ext_vector_type(8)));

#define B_DIM 128
#define M_DIM 2048
#define D_DIM 1024
#define P_DIM 64
#define BETA_F 10.0f
#define INV_SCALE 0.125f /* 1/sqrt(64) */

#define MC 16
#define NCHUNK (M_DIM / MC)                 /* 128 */
#define CHUNK_BYTES (MC * D_DIM * 4)        /* 65536 */
#define CHUNK_ELEMS (MC * D_DIM)            /* 16384 fp32 */

// ---------------------------------------------------------------------------
// Kernel A: T = query @ Wq^T   (128 x 64, K = 1024), fp32 WMMA 16x16x4
// ---------------------------------------------------------------------------
__global__ __launch_bounds__(256) void proj_q_kernel(const float* __restrict__ q,
                                                     const float* __restrict__ wq,
                                                     float* __restrict__ T) {
  const int lane = threadIdx.x & 31;
  const int wg   = blockIdx.x * 8 + (threadIdx.x >> 5); // 0..31
  const int tileM = wg >> 2;                            // 0..7
  const int tileN = wg & 3;                             // 0..3
  const int half = lane >> 4;
  const int l16  = lane & 15;
  const int mrow = tileM * 16 + l16;
  const int nrow = tileN * 16 + l16;
  v8f c = {};
  for (int k0 = 0; k0 < D_DIM; k0 += 4) {
    const int ka = k0 + half * 2;
    v2f a = { q[mrow * D_DIM + ka],  q[mrow * D_DIM + ka + 1] };
    v2f b = { wq[nrow * D_DIM + ka], wq[nrow * D_DIM + ka + 1] };
    c = __builtin_amdgcn_wmma_f32_16x16x4_f32(false, a, false, b,
                                              (short)0, c, false, false);
  }
  #pragma unroll
  for (int r = 0; r < 8; ++r) {
    const int m = tileM * 16 + half * 8 + r;
    T[m * P_DIM + tileN * 16 + l16] = c[r];
  }
}

// ---------------------------------------------------------------------------
// Kernel B: U = T @ Wk   (128 x 1024, K = 64), fp32 WMMA 16x16x4
// ---------------------------------------------------------------------------
__global__ __launch_bounds__(256) void proj_u_kernel(const float* __restrict__ T,
                                                     const float* __restrict__ wk,
                                                     float* __restrict__ U) {
  const int lane = threadIdx.x & 31;
  const int wg   = blockIdx.x * 8 + (threadIdx.x >> 5); // 0..511
  const int tileM = wg >> 6;                            // 0..7
  const int tileN = wg & 63;                            // 0..63
  const int half = lane >> 4;
  const int l16  = lane & 15;
  const int mrow = tileM * 16 + l16;
  const int ncol = tileN * 16 + l16;
  v8f c = {};
  #pragma unroll
  for (int k0 = 0; k0 < P_DIM; k0 += 4) {
    const int ka = k0 + half * 2;
    v2f a = { T[mrow * P_DIM + ka],  T[mrow * P_DIM + ka + 1] };
    v2f b = { wk[ka * D_DIM + ncol], wk[(ka + 1) * D_DIM + ncol] };
    c = __builtin_amdgcn_wmma_f32_16x16x4_f32(false, a, false, b,
                                              (short)0, c, false, false);
  }
  #pragma unroll
  for (int r = 0; r < 8; ++r) {
    const int m = tileM * 16 + half * 8 + r;
    U[m * D_DIM + tileN * 16 + l16] = c[r];
  }
}

// ---------------------------------------------------------------------------
// TDM descriptor helper: 1-D tile of CHUNK_ELEMS fp32 (64 KB), D# per ISA §8.3-8.6
//   group0: count=1 | lds_addr | global_addr[56:0] | type=2
//   group1: data_size=2(4B); tensor_dim0=tile_dim0=CHUNK_ELEMS; tensor_dim1=1
// 6-arg builtin variant (clang-23 / therock-10.0): extra int32x8 group + cpol.
// ---------------------------------------------------------------------------
__device__ __forceinline__ void tdm_load_chunk(unsigned ldsAddr, unsigned long long ga) {
  v4u g0;
  g0[0] = 1u;                                   // count=1, user mode
  g0[1] = ldsAddr;                              // lds_addr
  g0[2] = (unsigned)(ga & 0xFFFFFFFFull);       // global_addr[31:0]
  g0[3] = (unsigned)((ga >> 32) & 0x1FFFFFFull) // global_addr[56:32]
          | 0x80000000u;                        // type=2 (bits 127:126)
  v8i g1;
  g1[0] = 0x20000;                              // data_size=2 (4 bytes), mask=0
  g1[1] = (int)((CHUNK_ELEMS & 0xFFFF) << 16);  // tensor_dim0[15:0] @ bits 63:48
  g1[2] = (int)((CHUNK_ELEMS >> 16) |           // tensor_dim0[31:16]
                (1u << 16));                    // tensor_dim1 = 1 (bits 95:80)
  g1[3] = (int)((CHUNK_ELEMS & 0xFFFF) << 16);  // tile_dim0 @ bits 127:112 (td1 hi=0)
  g1[4] = 0;                                    // tile_dim1=0, tile_dim2=0 (unused)
  g1[5] = (int)CHUNK_ELEMS;                     // tensor_dim0_stride[31:0]
  g1[6] = 0;
  g1[7] = 0;
  v4i gz4 = {0, 0, 0, 0};
  v8i gz8 = {0, 0, 0, 0, 0, 0, 0, 0};
  __builtin_amdgcn_tensor_load_to_lds(g0, g1, gz4, gz4, gz8, 0);
}

// ---------------------------------------------------------------------------
// Kernel C: streaming fused hopfield pass. One block (8 waves) per batch row.
// TDM double-buffers 16-row (64 KB) candidate chunks into LDS; flash-style
// online softmax + fused readout (single HBM pass over 1 GiB).
// ---------------------------------------------------------------------------
__global__ __launch_bounds__(256) void hopfield_stream_kernel(
    const float* __restrict__ cand, const float* __restrict__ U,
    float* __restrict__ out) {
  extern __shared__ float lds[];   // 2 * MC * D_DIM floats (128 KB)
  __shared__ float sScore[MC];

  const int tid  = threadIdx.x;
  const int lane = tid & 31;
  const int wave = tid >> 5;
  const int b    = blockIdx.x;

  const float* candB = cand + (size_t)b * M_DIM * D_DIM;
  const unsigned long long gbase = (unsigned long long)(size_t)candB;
  const unsigned ldsBase = (unsigned)(size_t)(&lds[0]);

  // u[b] lane-sliced float4 registers: useg[s] = u[s*128 + lane*4 .. +3]
  float4 useg[8];
  #pragma unroll
  for (int s = 0; s < 8; ++s)
    useg[s] = *(const float4*)(U + b * D_DIM + s * 128 + lane * 4);

  float acc0 = 0.f, acc1 = 0.f, acc2 = 0.f, acc3 = 0.f;
  float mv = -INFINITY, sv = 0.f;
  const int d0 = tid * 4;

  // Prologue: wave 0 kicks off chunk 0 into buffer 0
  if (wave == 0) tdm_load_chunk(ldsBase, gbase);

  int buf = 0;
  for (int cth = 0; cth < NCHUNK; ++cth) {
    if (wave == 0) {
      if (cth + 1 < NCHUNK) {
        tdm_load_chunk(ldsBase + (unsigned)(buf ^ 1) * (unsigned)CHUNK_BYTES,
                       gbase + (unsigned long long)(cth + 1) * CHUNK_BYTES);
        __builtin_amdgcn_s_wait_tensorcnt(1);   // chunk cth resident (in-order)
      } else {
        __builtin_amdgcn_s_wait_tensorcnt(0);
      }
    }
    __syncthreads();

    const float* chunk = &lds[buf * (MC * D_DIM)];

    // scores: each wave computes rows 2w, 2w+1 (float4 LDS reads)
    #pragma unroll
    for (int rr = 0; rr < 2; ++rr) {
      const int row = wave * 2 + rr;
      const float4* crow = (const float4*)(chunk + row * D_DIM);
      float s = 0.f;
      #pragma unroll
      for (int seg = 0; seg < 8; ++seg) {
        const float4 cv = crow[seg * 32 + lane];
        s = fmaf(cv.x, useg[seg].x, s);
        s = fmaf(cv.y, useg[seg].y, s);
        s = fmaf(cv.z, useg[seg].z, s);
        s = fmaf(cv.w, useg[seg].w, s);
      }
      #pragma unroll
      for (int off = 16; off > 0; off >>= 1) s += __shfl_xor(s, off, 32);
      if (lane == 0) sScore[row] = s * (BETA_F * INV_SCALE); // t = beta*score
    }
    __syncthreads();

    // online softmax update (identical scalar path in all threads)
    float smax = sScore[0];
    #pragma unroll
    for (int j = 1; j < MC; ++j) smax = fmaxf(smax, sScore[j]);
    const float newmax = fmaxf(mv, smax);
    const float factor = __expf(mv - newmax);
    float p[MC], psum = 0.f;
    #pragma unroll
    for (int j = 0; j < MC; ++j) { p[j] = __expf(sScore[j] - newmax); psum += p[j]; }
    acc0 *= factor; acc1 *= factor; acc2 *= factor; acc3 *= factor;
    #pragma unroll
    for (int j = 0; j < MC; ++j) {
      const float4 cv = *(const float4*)(chunk + j * D_DIM + d0);
      acc0 = fmaf(p[j], cv.x, acc0);
      acc1 = fmaf(p[j], cv.y, acc1);
      acc2 = fmaf(p[j], cv.z, acc2);
      acc3 = fmaf(p[j], cv.w, acc3);
    }
    sv = sv * factor + psum;
    mv = newmax;
    __syncthreads();   // all readers done before buf is overwritten next round
    buf ^= 1;
  }

  const float inv = 1.0f / sv;
  float* ro = out + (size_t)b * D_DIM + d0;
  ro[0] = acc0 * inv; ro[1] = acc1 * inv; ro[2] = acc2 * inv; ro[3] = acc3 * inv;
  if (tid == 0)
    out[(size_t)B_DIM * D_DIM + b] = -((mv + __logf(sv)) / BETA_F);
}

// ---------------------------------------------------------------------------
extern "C" void kernel_launch(void* const* d_in, const int* in_sizes, int n_in,
                              void* d_out, int out_size, void* d_ws, size_t ws_size,
                              hipStream_t stream) {
  (void)in_sizes; (void)n_in; (void)out_size; (void)ws_size;
  const float* q    = (const float*)d_in[0];
  const float* cand = (const float*)d_in[1];
  const float* wq   = (const float*)d_in[2];
  const float* wk   = (const float*)d_in[3];
  float* out = (float*)d_out;

  float* U = (float*)d_ws;                                                  // 512 KB
  float* T = (float*)((char*)d_ws + (size_t)B_DIM * D_DIM * sizeof(float)); // 32 KB

  proj_q_kernel<<<4, 256, 0, stream>>>(q, wq, T);
  proj_u_kernel<<<64, 256, 0, stream>>>(T, wk, U);
  hopfield_stream_kernel<<<B_DIM, 256, 2 * MC * D_DIM * sizeof(float), stream>>>(cand, U, out);
}